// CXLMultiHeadEmbedding_25683904430107
// MI455X (gfx1250) — compile-verified
//
#include <hip/hip_runtime.h>
#include <cstdint>

// CXL multi-head embedding gather for MI455X (gfx1250).
//
// out[row, 0:64] = table[ids[row] + (row%8)*100000, 0:64]
//   rows = B*L*H = 524288, row bytes = 256 (64 x f32)
//
// Memory-bound: ~264 MiB total traffic -> ~12us floor at 23.3 TB/s.
// Strategy: CDNA5 async global<->LDS DMA path. Each wave gathers 32 rows
// (8 KB) into a private LDS window with global_load_async_to_lds_b128
// (each instruction = 2 full contiguous 256B rows across the wave; 16
// in flight per wave), then drains the window to the contiguous output
// slice with global_store_async_from_lds_b128 (512B per instruction,
// th:TH_STORE_NT so the write-once output stream does not evict table
// rows from the 192MB L2). Data never touches the VGPR file; ASYNCcnt
// tracks completion.

#define WAVES_PER_BLOCK 8
#define ROWS_PER_WAVE   32          // rows staged per wave per iteration
#define ROW_BYTES       256         // 64 floats
#define LDS_PER_WAVE    (ROWS_PER_WAVE * ROW_BYTES)   // 8 KB
#define HEAD_N          100000u     // rows per head segment (LIST_OF_N[i])

__global__ __launch_bounds__(256) void cxl_embed_gather_async(
    const int*   __restrict__ ids,    // [rows] flattened [B,L,H]
    const float* __restrict__ table,  // [800000, 64]
    float*       __restrict__ out,    // [rows, 64]
    int nRows)
{
    __shared__ unsigned char smem[WAVES_PER_BLOCK * LDS_PER_WAVE];

    const int lane  = threadIdx.x & 31;
    const int wave  = threadIdx.x >> 5;
    const int chunk = lane & 15;   // 16-byte chunk id within a 256B row
    const int half  = lane >> 4;   // 0 -> even row of pair, 1 -> odd row

    // 32-bit LDS byte offset of this wave's staging window. Flat shared
    // pointers carry the LDS aperture in the high 32 bits (ISA 10.2), so
    // truncation yields the workgroup-relative LDS address the async ops want.
    const uint32_t ldsWaveBase =
        (uint32_t)(uintptr_t)(&smem[wave * LDS_PER_WAVE]);

    const uint64_t tableBase = (uint64_t)(uintptr_t)table;  // SGPR pair (GVS)
    const uint64_t outBase   = (uint64_t)(uintptr_t)out;    // SGPR pair (GVS)

    const int rowStride = (int)gridDim.x * WAVES_PER_BLOCK * ROWS_PER_WAVE;

    for (int rowBase = ((int)blockIdx.x * WAVES_PER_BLOCK + wave) * ROWS_PER_WAVE;
         rowBase < nRows; rowBase += rowStride)
    {
        // Lane l owns row rowBase+l. rowBase is a multiple of 32 and H == 8,
        // so head(row) = lane & 7; per-head table offset = head * 100000.
        uint32_t rowIdx = (uint32_t)ids[rowBase + lane]
                        + (uint32_t)(lane & 7) * HEAD_N;

        // ---- Stage 1: gather 32 rows into LDS (16 async ops, 512B each) ----
        #pragma unroll
        for (int j = 0; j < 16; ++j) {
            // Broadcast row index of row (rowBase + 2j + half) to this lane.
            uint32_t r = (uint32_t)__shfl((int)rowIdx, 2 * j + half, 32);
            uint32_t goff  = (r << 8) + (uint32_t)(chunk << 4);              // byte off into table
            uint32_t laddr = ldsWaveBase + (uint32_t)(j * 512 + lane * 16);  // LDS dest
            asm volatile("global_load_async_to_lds_b128 %0, %1, %2"
                         :
                         : "v"(laddr), "v"(goff), "s"(tableBase)
                         : "memory");
        }
        asm volatile("s_wait_asynccnt 0" ::: "memory");   // LDS window filled

        // ---- Stage 2: drain 8KB LDS window to contiguous output (NT) ----
        const uint32_t outOff0 = (uint32_t)rowBase * ROW_BYTES + (uint32_t)(lane * 16);
        #pragma unroll
        for (int k = 0; k < 16; ++k) {
            uint32_t laddr = ldsWaveBase + (uint32_t)(k * 512 + lane * 16);
            uint32_t ooff  = outOff0 + (uint32_t)(k * 512);
            asm volatile("global_store_async_from_lds_b128 %0, %1, %2 th:TH_STORE_NT"
                         :
                         : "v"(ooff), "v"(laddr), "s"(outBase)
                         : "memory");
        }
        asm volatile("s_wait_asynccnt 0" ::: "memory");   // window reusable
    }
}

extern "C" void kernel_launch(void* const* d_in, const int* in_sizes, int n_in,
                              void* d_out, int out_size, void* d_ws, size_t ws_size,
                              hipStream_t stream)
{
    (void)in_sizes; (void)n_in; (void)d_ws; (void)ws_size;

    const int*   ids   = (const int*)d_in[0];    // input_ids, flat [B,L,H]
    const float* table = (const float*)d_in[1];  // [800000, 64]
    float*       outp  = (float*)d_out;          // [B,L,H,64]

    const int nRows = out_size / 64;             // 524288 for reference shapes

    int blocks = nRows / (WAVES_PER_BLOCK * ROWS_PER_WAVE);  // 2048
    if (blocks < 1)     blocks = 1;
    if (blocks > 65535) blocks = 65535;          // grid-stride loop covers rest

    cxl_embed_gather_async<<<blocks, 256, 0, stream>>>(ids, table, outp, nRows);
}